// NNAttention_5669356832198
// MI455X (gfx1250) — compile-verified
//
#include <hip/hip_runtime.h>

// Problem constants (match reference)
constexpr int kB   = 32;
constexpr int kT   = 2048;
constexpr int kH   = 512;
constexpr int kK   = 15;   // conv taps
constexpr int kPad = 7;    // (kK-1)/2

typedef __attribute__((ext_vector_type(2))) float v2f;
typedef __attribute__((ext_vector_type(8))) float v8f;

// ---------------------------------------------------------------------------
// Kernel 1: fused conv (via f32 WMMA) + pax + relu + dot(lin_w) -> scores[b,t]
// grid = (kT/128, kB), block = 256 (8 waves); wave w owns t-tile t0 = bx*128+w*16
// ---------------------------------------------------------------------------
__global__ __launch_bounds__(256) void scores_kernel(
    const float* __restrict__ eh,      // [B,T,H]
    const float* __restrict__ dhx,     // [B,H]
    const float* __restrict__ ax,      // [B,T]
    const float* __restrict__ conv_w,  // [H,1,K] flat = h*kK + k
    const float* __restrict__ conv_b,  // [H]
    const float* __restrict__ lin_w,   // [H]
    const float* __restrict__ lin_b,   // [1]
    float* __restrict__ scores)        // [B,T] (attn region of d_out)
{
  const int b    = blockIdx.y;
  const int wave = threadIdx.x >> 5;
  const int lane = threadIdx.x & 31;
  const int half = lane >> 4;   // 0 => lanes 0-15, 1 => lanes 16-31
  const int l16  = lane & 15;
  const int t0   = blockIdx.x * 128 + wave * 16;

  const float* axb = ax + b * kT;

  // A fragments (16x4 f32, M = t-in-tile, K = conv tap), one per K-chunk.
  // VGPR0: lanes0-15 K=kc*4+0 / lanes16-31 K=kc*4+2 ; VGPR1: +1 / +3.
  // A[m,k] = ax[b, t0 + m + k - kPad], zero outside [0,T) or k >= kK.
  v2f afrag[4];
  const int m = l16;
#pragma unroll
  for (int kc = 0; kc < 4; ++kc) {
    const int kbase = kc * 4 + half * 2;
    const int tt0 = t0 + m + kbase - kPad;
    float a0 = 0.0f, a1 = 0.0f;
    if (kbase     < kK && tt0     >= 0 && tt0     < kT) a0 = axb[tt0];
    if (kbase + 1 < kK && tt0 + 1 >= 0 && tt0 + 1 < kT) a1 = axb[tt0 + 1];
    afrag[kc].x = a0;
    afrag[kc].y = a1;
  }

  float acc[8];
#pragma unroll
  for (int j = 0; j < 8; ++j) acc[j] = 0.0f;

  const float* ehb  = eh  + (size_t)b * kT * kH;
  const float* dhxb = dhx + b * kH;

  for (int ht = 0; ht < kH / 16; ++ht) {
    const int h = ht * 16 + l16;   // N = lane&15

    // B fragment (4x16 f32): B[k,n] = conv_w[h0+n, k]; same half-wave K split.
    v8f c = {};
#pragma unroll
    for (int kc = 0; kc < 4; ++kc) {
      const int kbase = kc * 4 + half * 2;
      v2f bfrag;
      bfrag.x = (kbase     < kK) ? conv_w[h * kK + kbase]     : 0.0f;
      bfrag.y = (kbase + 1 < kK) ? conv_w[h * kK + kbase + 1] : 0.0f;
      c = __builtin_amdgcn_wmma_f32_16x16x4_f32(
          /*neg_a=*/false, afrag[kc], /*neg_b=*/false, bfrag,
          /*c_mod=*/(short)0, c, /*reuse_a=*/false, /*reuse_b=*/false);
    }

    const float d  = dhxb[h];
    const float cb = conv_b[h];
    const float lw = lin_w[h];

    // D layout: VGPR j -> M = j + 8*half, N = lane&15
#pragma unroll
    for (int j = 0; j < 8; ++j) {
      const int t = t0 + j + half * 8;
      const float pax = ehb[(size_t)t * kH + h] + d + cb + c[j];
      acc[j] += fmaxf(pax, 0.0f) * lw;
    }
  }

  const float lb = lin_b[0];
#pragma unroll
  for (int j = 0; j < 8; ++j) {
    float v = acc[j];
#pragma unroll
    for (int off = 8; off >= 1; off >>= 1)
      v += __shfl_xor(v, off, 16);          // reduce within each half-wave (N dim)
    if (l16 == 0)
      scores[b * kT + t0 + j + half * 8] = v + lb;
  }
}

// ---------------------------------------------------------------------------
// Kernel 2: softmax over T, in place in the attn region. grid = kB, block = 256
// ---------------------------------------------------------------------------
__global__ __launch_bounds__(256) void softmax_kernel(float* __restrict__ attn)
{
  __shared__ float s[kT];      // 8 KB
  __shared__ float red[256];
  const int b   = blockIdx.x;
  const int tid = threadIdx.x;
  float* row = attn + b * kT;

  float lmax = -INFINITY;
  for (int t = tid; t < kT; t += 256) {
    const float v = row[t];
    s[t] = v;
    lmax = fmaxf(lmax, v);
  }
  red[tid] = lmax;
  __syncthreads();
  for (int off = 128; off >= 1; off >>= 1) {
    if (tid < off) red[tid] = fmaxf(red[tid], red[tid + off]);
    __syncthreads();
  }
  const float mx = red[0];
  __syncthreads();

  float lsum = 0.0f;
  for (int t = tid; t < kT; t += 256) {
    const float e = __expf(s[t] - mx);
    s[t] = e;
    lsum += e;
  }
  red[tid] = lsum;
  __syncthreads();
  for (int off = 128; off >= 1; off >>= 1) {
    if (tid < off) red[tid] += red[tid + off];
    __syncthreads();
  }
  const float inv = 1.0f / red[0];
  for (int t = tid; t < kT; t += 256) row[t] = s[t] * inv;
}

// ---------------------------------------------------------------------------
// Kernel 3: sx[b,h] = sum_t eh[b,t,h] * attn[b,t]
// grid = (kH/256, kB), block = 256; attn tiles staged through LDS
// ---------------------------------------------------------------------------
__global__ __launch_bounds__(256) void sx_kernel(
    const float* __restrict__ eh,    // [B,T,H]
    const float* __restrict__ attn,  // [B,T]
    float* __restrict__ sx)          // [B,H]
{
  __shared__ float at[256];
  const int b = blockIdx.y;
  const int h = blockIdx.x * 256 + threadIdx.x;
  const float* ehb = eh + (size_t)b * kT * kH;
  const float* ab  = attn + b * kT;

  float acc = 0.0f;
  for (int tt = 0; tt < kT; tt += 256) {
    __syncthreads();
    at[threadIdx.x] = ab[tt + threadIdx.x];
    __syncthreads();
    if (tt + 256 < kT)  // prefetch next eh tile (global_prefetch_b8)
      __builtin_prefetch(&ehb[(size_t)(tt + 256) * kH + h], 0, 0);
#pragma unroll 8
    for (int k = 0; k < 256; ++k)
      acc += ehb[(size_t)(tt + k) * kH + h] * at[k];
  }
  sx[b * kH + h] = acc;
}

// ---------------------------------------------------------------------------
extern "C" void kernel_launch(void* const* d_in, const int* in_sizes, int n_in,
                              void* d_out, int out_size, void* d_ws, size_t ws_size,
                              hipStream_t stream) {
  const float* eh     = (const float*)d_in[0];
  const float* dhx    = (const float*)d_in[1];
  const float* ax     = (const float*)d_in[2];
  const float* conv_w = (const float*)d_in[3];
  const float* conv_b = (const float*)d_in[4];
  const float* lin_w  = (const float*)d_in[5];
  const float* lin_b  = (const float*)d_in[6];

  float* sx   = (float*)d_out;             // [B,H] first
  float* attn = (float*)d_out + kB * kH;   // [B,T] second (scores staged here)

  dim3 g1(kT / 128, kB);
  scores_kernel<<<g1, 256, 0, stream>>>(eh, dhx, ax, conv_w, conv_b,
                                        lin_w, lin_b, attn);
  softmax_kernel<<<kB, 256, 0, stream>>>(attn);
  dim3 g3(kH / 256, kB);
  sx_kernel<<<g3, 256, 0, stream>>>(eh, attn, sx);
}